// InferModel_75230647156851
// MI455X (gfx1250) — compile-verified
//
#include <hip/hip_runtime.h>

typedef float v2f __attribute__((ext_vector_type(2)));
typedef float v8f __attribute__((ext_vector_type(8)));

#define SEM_W   256
#define SEM_H   288
#define SEM_C   10
#define CAM_STRIDE (SEM_C * SEM_H * SEM_W)   // 737280
#define CH_STRIDE  (SEM_H * SEM_W)           // 73728
#define HM_W    640
#define HM_N    (HM_W * HM_W)                // 409600
#define MAXDET  15

// ---------------------------------------------------------------------------
// Fuse K * perm * W2C_i * L2W into one 3x4 matrix per camera (36 floats).
// ---------------------------------------------------------------------------
__global__ void prep_P(const float* __restrict__ K, const float* __restrict__ L2W,
                       const float* __restrict__ W2C, float* __restrict__ P) {
    if (threadIdx.x != 0 || blockIdx.x != 0) return;
    for (int cam = 0; cam < 3; ++cam) {
        const float* W = W2C + cam * 16;
        float M[4][4];
        for (int i = 0; i < 4; ++i)
            for (int j = 0; j < 4; ++j) {
                float s = 0.f;
                for (int k = 0; k < 4; ++k) s += W[i * 4 + k] * L2W[k * 4 + j];
                M[i][j] = s;
            }
        // cam3 rows = [cam[1], -cam[2], cam[0]]
        float C3[3][4];
        for (int j = 0; j < 4; ++j) {
            C3[0][j] =  M[1][j];
            C3[1][j] = -M[2][j];
            C3[2][j] =  M[0][j];
        }
        for (int r = 0; r < 3; ++r)
            for (int c = 0; c < 4; ++c)
                P[cam * 12 + r * 4 + c] =
                    K[r * 3 + 0] * C3[0][c] + K[r * 3 + 1] * C3[1][c] + K[r * 3 + 2] * C3[2][c];
    }
}

// ---------------------------------------------------------------------------
// Point painting: one wave32 per block, 32 points per group iteration.
// Projection done with V_WMMA_F32_16X16X4_F32 (two 16-point tiles/group).
// ---------------------------------------------------------------------------
__global__ __launch_bounds__(32) void paint_kernel(
    const float* __restrict__ lidar, const float* __restrict__ sems,
    const float* __restrict__ P, float* __restrict__ out, int nGroups) {

    __shared__ float ldsD[2 * 16 * 16];   // D tiles: [tile][M=point][N=component]
    __shared__ float ldsOut[32 * SEM_C];  // staged painted rows for coalesced store

    const int lane = threadIdx.x;         // 0..31
    const int n    = lane & 15;           // N column / component id
    const int hi   = lane >> 4;           // half-wave select (K=0,1 vs K=2,3)

    // B matrix (4x16 f32): column n holds P[cam=n/3][row=n%3][k], zero for n>=9.
    // Layout mirrors the ISA 16x4 A table: lanes 0-15 -> K=0(,1), lanes 16-31 -> K=2(,3).
    v2f b = {0.f, 0.f};
    if (n < 9) {
        const int k0 = hi * 2;
        b.x = P[n * 4 + k0];
        b.y = P[n * 4 + k0 + 1];
    }

    for (int g = blockIdx.x; g < nGroups; g += gridDim.x) {
        const int gn = g + gridDim.x;
        if (gn < nGroups)  // warm L2 for the next group's lidar block (global_prefetch_b8)
            __builtin_prefetch((const void*)(lidar + (size_t)gn * 128 + lane * 4), 0, 1);

        // ---- phase 1: project 2 tiles of 16 points via WMMA ----
        #pragma unroll
        for (int t = 0; t < 2; ++t) {
            const int p = g * 32 + t * 16 + n;        // point this lane-pair loads
            // lanes 0-15: (x,y) ; lanes 16-31: (z, w) with w forced to 1.0
            v2f a = *reinterpret_cast<const v2f*>(lidar + (size_t)p * 4 + hi * 2);
            if (hi) a.y = 1.0f;
            v8f acc = {};
            acc = __builtin_amdgcn_wmma_f32_16x16x4_f32(
                false, a, false, b, (short)0, acc, false, false);
            // D: VGPR r holds M=r (lanes 0-15) and M=r+8 (lanes 16-31), N=lane%16
            float* ld = &ldsD[t * 256];
            #pragma unroll
            for (int r = 0; r < 8; ++r)
                ld[(r + hi * 8) * 16 + n] = acc[r];
        }
        __syncthreads();   // single-wave workgroup: compiles to S_NOP, orders LDS

        // ---- phase 2: each lane owns one point; divide, validate, gather ----
        const float* row = &ldsD[(lane >> 4) * 256 + (lane & 15) * 16];
        float vals[9];
        #pragma unroll
        for (int j = 0; j < 9; ++j) vals[j] = row[j];

        int sel = -1, su = 0, sv = 0;
        #pragma unroll
        for (int cam = 0; cam < 3; ++cam) {
            const float cx = vals[cam * 3 + 0];
            const float cy = vals[cam * 3 + 1];
            const float cz = vals[cam * 3 + 2];
            const float zz = cz + 1e-5f;
            const int u  = (int)(cx / zz);
            const int v  = (int)(cy / zz);
            const int zi = (int)cz;
            if (zi >= 0 && u >= 0 && u < SEM_W && v >= 0 && v < SEM_H) {
                sel = cam; su = u; sv = v;   // later cameras overwrite (matches ref)
            }
        }

        if (sel >= 0) {
            const float* basep = sems + (size_t)sel * CAM_STRIDE + sv * SEM_W + su;
            #pragma unroll
            for (int c = 0; c < SEM_C; ++c)
                ldsOut[lane * SEM_C + c] = basep[(size_t)c * CH_STRIDE];
        } else {
            #pragma unroll
            for (int c = 0; c < SEM_C; ++c)
                ldsOut[lane * SEM_C + c] = 0.0f;
        }
        __syncthreads();

        // ---- phase 3: fully coalesced 1280B store of the group's painted rows ----
        float* outp = out + (size_t)g * (32 * SEM_C);
        #pragma unroll
        for (int j = 0; j < SEM_C; ++j)
            outp[j * 32 + lane] = ldsOut[j * 32 + lane];
        __syncthreads();
    }
}

// ---------------------------------------------------------------------------
// Heatmap peaks: separable 7-max, then 15 argmax passes per map.
// ---------------------------------------------------------------------------
__global__ void rowmax7(const float* __restrict__ hm, float* __restrict__ tmp) {
    const int idx = blockIdx.x * blockDim.x + threadIdx.x;
    if (idx >= 2 * HM_N) return;
    const int x  = idx % HM_W;
    const int yh = idx / HM_W;                 // h*640 + y
    const int x0 = max(x - 3, 0), x1 = min(x + 3, HM_W - 1);
    const float* r = hm + (size_t)yh * HM_W;
    float m = r[x0];
    for (int xx = x0 + 1; xx <= x1; ++xx) m = fmaxf(m, r[xx]);
    tmp[idx] = m;
}

__global__ void colposs7(const float* __restrict__ hm, const float* __restrict__ tmp,
                         float* __restrict__ poss) {
    const int idx = blockIdx.x * blockDim.x + threadIdx.x;
    if (idx >= 2 * HM_N) return;
    const int x = idx % HM_W;
    const int y = (idx / HM_W) % HM_W;
    const int h = idx / HM_N;
    const int y0 = max(y - 3, 0), y1 = min(y + 3, HM_W - 1);
    const float* col = tmp + (size_t)h * HM_N;
    float m = col[y0 * HM_W + x];
    for (int yy = y0 + 1; yy <= y1; ++yy) m = fmaxf(m, col[yy * HM_W + x]);
    const float v = hm[idx];
    poss[idx] = v - ((m > v) ? 100000.0f : 0.0f);
}

__global__ void topk15(float* __restrict__ poss, float* __restrict__ scores,
                       float* __restrict__ locs) {
    __shared__ float svv[256];
    __shared__ int   sll[256];
    const int h = blockIdx.x;
    float* p = poss + (size_t)h * HM_N;
    const int tid = threadIdx.x;

    for (int pass = 0; pass < MAXDET; ++pass) {
        float best = -3.4e38f;
        int   bloc = 0x7fffffff;
        for (int i = tid; i < HM_N; i += 256) {
            const float s = p[i];
            if (s > best || (s == best && i < bloc)) { best = s; bloc = i; }
        }
        svv[tid] = best; sll[tid] = bloc;
        __syncthreads();
        for (int off = 128; off > 0; off >>= 1) {
            if (tid < off) {
                const float s2 = svv[tid + off];
                const int   l2 = sll[tid + off];
                if (s2 > svv[tid] || (s2 == svv[tid] && l2 < sll[tid])) {
                    svv[tid] = s2; sll[tid] = l2;
                }
            }
            __syncthreads();
        }
        if (tid == 0) {
            scores[h * MAXDET + pass] = svv[0];
            locs[h * MAXDET + pass]   = (float)sll[0];  // int loc promoted to f32 in concat
            p[sll[0]] = -3.4e38f;                       // remove winner for next pass
        }
        __threadfence();
        __syncthreads();
    }
}

// ---------------------------------------------------------------------------
extern "C" void kernel_launch(void* const* d_in, const int* in_sizes, int n_in,
                              void* d_out, int out_size, void* d_ws, size_t ws_size,
                              hipStream_t stream) {
    const float* lidar = (const float*)d_in[0];   // (N,4)
    const float* sems  = (const float*)d_in[1];   // (3,10,288,256)
    const float* hm    = (const float*)d_in[2];   // (2,640,640)
    const float* Km    = (const float*)d_in[3];   // (3,3)
    const float* L2W   = (const float*)d_in[4];   // (4,4)
    const float* W2C   = (const float*)d_in[5];   // (3,4,4)
    float* out = (float*)d_out;
    float* ws  = (float*)d_ws;

    float* P    = ws;                    // 36 floats (pad to 64)
    float* tmp  = ws + 64;               // 819200 floats
    float* poss = ws + 64 + 2 * HM_N;    // 819200 floats

    const int nPoints = in_sizes[0] / 4; // 2,000,000
    const int nGroups = nPoints / 32;    // 62,500 (exact)

    prep_P<<<1, 32, 0, stream>>>(Km, L2W, W2C, P);
    paint_kernel<<<4096, 32, 0, stream>>>(lidar, sems, P, out, nGroups);

    const int hmTotal = 2 * HM_N;
    rowmax7<<<(hmTotal + 255) / 256, 256, 0, stream>>>(hm, tmp);
    colposs7<<<(hmTotal + 255) / 256, 256, 0, stream>>>(hm, tmp, poss);

    float* scores = out + (size_t)nPoints * SEM_C;       // 2x15
    float* locs   = scores + 2 * MAXDET;                 // 2x15
    topk15<<<2, 256, 0, stream>>>(poss, scores, locs);
}